// GNNLayer_75196287418454
// MI455X (gfx1250) — compile-verified
//
#include <hip/hip_runtime.h>
#include <hip/hip_bf16.h>
#include <math.h>

#define N_NODES 50000
#define N_EDGES 800000
#define ETOT    (N_EDGES + N_NODES)
#define C       128
#define BN_EPS  1e-5f

typedef __attribute__((ext_vector_type(2))) float v2f;
typedef __attribute__((ext_vector_type(8))) float v8f;

// ---------------------------------------------------------------------------
// Init: zero d_out (poisoned 0xAA by harness), zero denom, set amax = -inf
// ---------------------------------------------------------------------------
__global__ __launch_bounds__(256) void init_kernel(float* __restrict__ out,
                                                   float* __restrict__ amax,
                                                   float* __restrict__ denom) {
    int i = blockIdx.x * blockDim.x + threadIdx.x;
    if (i < N_NODES * C) out[i] = 0.0f;
    if (i < N_NODES) {
        amax[i]  = -INFINITY;
        denom[i] = 0.0f;
    }
}

// ---------------------------------------------------------------------------
// GEMM: xl[N,128] = x[N,128] @ W[128,128]^T   via V_WMMA_F32_16X16X4_F32
// Block = 256 threads = 8 waves; wave w computes 16x16 tile at (blockIdx*16, w*16)
// A(16x4 f32): lane = M + 16*(K>=2), vgpr = K%2. B(4x16): lane = N + 16*(K>=2).
// C(16x16): vgpr r -> rows r (lanes 0-15) and r+8 (lanes 16-31), N = lane%16.
// ---------------------------------------------------------------------------
__global__ __launch_bounds__(256) void gemm_xl_kernel(const float* __restrict__ x,
                                                      const float* __restrict__ W,
                                                      float* __restrict__ xl) {
    const int wave  = threadIdx.x >> 5;
    const int lane  = threadIdx.x & 31;
    const int tileM = blockIdx.x * 16;
    const int tileN = wave * 16;

    const int mn    = lane & 15;   // M index for A, N index for B
    const int khalf = lane >> 4;   // 0 -> K{0,1}, 1 -> K{2,3}

    const float* arow = x + (size_t)(tileM + mn) * C;  // x[m][k]
    const float* brow = W + (size_t)(tileN + mn) * C;  // W[n][k] == B[k][n]

    v8f c = {};
    #pragma unroll
    for (int k = 0; k < C; k += 4) {
        v2f a = *(const v2f*)(arow + k + 2 * khalf);
        v2f b = *(const v2f*)(brow + k + 2 * khalf);
        c = __builtin_amdgcn_wmma_f32_16x16x4_f32(
                /*neg_a=*/false, a, /*neg_b=*/false, b,
                /*c_mod=*/(short)0, c, /*reuse_a=*/false, /*reuse_b=*/false);
    }

    const int col = tileN + mn;
    #pragma unroll
    for (int r = 0; r < 8; ++r) {
        int row = tileM + r + 8 * khalf;
        xl[(size_t)row * C + col] = c[r];
    }
}

// ---------------------------------------------------------------------------
// Float atomic max via int punning (amax initialized to -inf)
// ---------------------------------------------------------------------------
__device__ __forceinline__ void atomic_max_f32(float* addr, float val) {
    if (val >= 0.0f)
        atomicMax((int*)addr, __float_as_int(val));
    else
        atomicMin((unsigned int*)addr, __float_as_uint(val));
}

// ---------------------------------------------------------------------------
// Per-edge attention logits + segment max. One wave per edge (32 x float4 = 128).
// Edges e >= N_EDGES are the appended self-loops (src = dst = e - N_EDGES).
// ---------------------------------------------------------------------------
__global__ __launch_bounds__(256) void edge_alpha_kernel(const int* __restrict__ esrc,
                                                         const int* __restrict__ edst,
                                                         const float* __restrict__ emb,
                                                         float* __restrict__ alpha,
                                                         float* __restrict__ amax) {
    const int e    = blockIdx.x * (blockDim.x >> 5) + (threadIdx.x >> 5);
    const int lane = threadIdx.x & 31;
    if (e >= ETOT) return;

    int s, d;
    if (e < N_EDGES) { s = esrc[e]; d = edst[e]; }
    else             { s = d = e - N_EDGES; }

    const float4 a = ((const float4*)(emb + (size_t)s * C))[lane];
    const float4 b = ((const float4*)(emb + (size_t)d * C))[lane];
    float p = a.x * b.x + a.y * b.y + a.z * b.z + a.w * b.w;

    #pragma unroll
    for (int off = 16; off > 0; off >>= 1) p += __shfl_down(p, off, 32);

    if (lane == 0) {
        alpha[e] = p;
        atomic_max_f32(&amax[d], p);
    }
}

// ---------------------------------------------------------------------------
// e = exp(alpha - amax[dst]); denom[dst] += e. One thread per edge.
// ---------------------------------------------------------------------------
__global__ __launch_bounds__(256) void edge_exp_kernel(const int* __restrict__ edst,
                                                       float* __restrict__ alpha,
                                                       const float* __restrict__ amax,
                                                       float* __restrict__ denom) {
    const int e = blockIdx.x * blockDim.x + threadIdx.x;
    if (e >= ETOT) return;
    const int d = (e < N_EDGES) ? edst[e] : (e - N_EDGES);
    const float ev = __expf(alpha[e] - amax[d]);
    alpha[e] = ev;
    atomicAdd(&denom[d], ev);
}

// ---------------------------------------------------------------------------
// out[dst] += xl[src] * (e / (denom[dst] + 1e-16)). One wave per edge.
// ---------------------------------------------------------------------------
__global__ __launch_bounds__(256) void aggregate_kernel(const int* __restrict__ esrc,
                                                        const int* __restrict__ edst,
                                                        const float* __restrict__ xl,
                                                        const float* __restrict__ alpha,
                                                        const float* __restrict__ denom,
                                                        float* __restrict__ out) {
    const int e    = blockIdx.x * (blockDim.x >> 5) + (threadIdx.x >> 5);
    const int lane = threadIdx.x & 31;
    if (e >= ETOT) return;

    int s, d;
    if (e < N_EDGES) { s = esrc[e]; d = edst[e]; }
    else             { s = d = e - N_EDGES; }

    const float a = alpha[e] / (denom[d] + 1e-16f);
    const float4 v = ((const float4*)(xl + (size_t)s * C))[lane];
    float* o = out + (size_t)d * C + lane * 4;
    atomicAdd(o + 0, v.x * a);
    atomicAdd(o + 1, v.y * a);
    atomicAdd(o + 2, v.z * a);
    atomicAdd(o + 3, v.w * a);
}

// ---------------------------------------------------------------------------
// Per-channel sum / sumsq of (out + bias). One block per channel.
// ---------------------------------------------------------------------------
__global__ __launch_bounds__(256) void stats_kernel(const float* __restrict__ out,
                                                    const float* __restrict__ bias,
                                                    float* __restrict__ chsum,
                                                    float* __restrict__ chsq) {
    const int c = blockIdx.x;
    const float b = bias[c];
    float s = 0.0f, q = 0.0f;
    for (int n = threadIdx.x; n < N_NODES; n += blockDim.x) {
        const float v = out[(size_t)n * C + c] + b;
        s += v;
        q += v * v;
    }
    __shared__ float ls[256], lq[256];
    ls[threadIdx.x] = s;
    lq[threadIdx.x] = q;
    __syncthreads();
    for (int off = 128; off > 0; off >>= 1) {
        if (threadIdx.x < off) {
            ls[threadIdx.x] += ls[threadIdx.x + off];
            lq[threadIdx.x] += lq[threadIdx.x + off];
        }
        __syncthreads();
    }
    if (threadIdx.x == 0) {
        chsum[c] = ls[0];
        chsq[c]  = lq[0];
    }
}

// ---------------------------------------------------------------------------
// BatchNorm (batch stats, biased var) + affine + ReLU, in place on d_out.
// ---------------------------------------------------------------------------
__global__ __launch_bounds__(256) void bn_relu_kernel(float* __restrict__ out,
                                                      const float* __restrict__ bias,
                                                      const float* __restrict__ gamma,
                                                      const float* __restrict__ beta,
                                                      const float* __restrict__ chsum,
                                                      const float* __restrict__ chsq) {
    const int i = blockIdx.x * blockDim.x + threadIdx.x;
    if (i >= N_NODES * C) return;
    const int c = i & (C - 1);
    const float inv_n = 1.0f / (float)N_NODES;
    const float mu  = chsum[c] * inv_n;
    const float var = chsq[c] * inv_n - mu * mu;
    const float v = (out[i] + bias[c] - mu) * rsqrtf(var + BN_EPS) * gamma[c] + beta[c];
    out[i] = fmaxf(v, 0.0f);
}

// ---------------------------------------------------------------------------
extern "C" void kernel_launch(void* const* d_in, const int* in_sizes, int n_in,
                              void* d_out, int out_size, void* d_ws, size_t ws_size,
                              hipStream_t stream) {
    (void)in_sizes; (void)n_in; (void)out_size; (void)ws_size;

    const float* x     = (const float*)d_in[0];
    const int*   edge  = (const int*)d_in[1];     // [2, N_EDGES]: row 0 = src, row 1 = dst
    const float* emb   = (const float*)d_in[2];
    const float* W     = (const float*)d_in[3];
    const float* bias  = (const float*)d_in[4];
    const float* gamma = (const float*)d_in[5];
    const float* beta  = (const float*)d_in[6];
    float*       out   = (float*)d_out;

    const int* esrc = edge;
    const int* edst = edge + N_EDGES;

    float* ws    = (float*)d_ws;
    float* xl    = ws;                       // N_NODES * C
    float* alpha = xl + (size_t)N_NODES * C; // ETOT
    float* amax  = alpha + ETOT;             // N_NODES
    float* denom = amax + N_NODES;           // N_NODES
    float* chsum = denom + N_NODES;          // C
    float* chsq  = chsum + C;                // C

    const int elems = N_NODES * C;

    init_kernel<<<(elems + 255) / 256, 256, 0, stream>>>(out, amax, denom);

    gemm_xl_kernel<<<N_NODES / 16, 256, 0, stream>>>(x, W, xl);

    // one wave per edge -> 8 edges per 256-thread block
    const int edge_wave_blocks = (ETOT + 7) / 8;
    edge_alpha_kernel<<<edge_wave_blocks, 256, 0, stream>>>(esrc, edst, emb, alpha, amax);

    edge_exp_kernel<<<(ETOT + 255) / 256, 256, 0, stream>>>(edst, alpha, amax, denom);

    aggregate_kernel<<<edge_wave_blocks, 256, 0, stream>>>(esrc, edst, xl, alpha, denom, out);

    stats_kernel<<<C, 256, 0, stream>>>(out, bias, chsum, chsq);

    bn_relu_kernel<<<(elems + 255) / 256, 256, 0, stream>>>(out, bias, gamma, beta, chsum, chsq);
}